// Revisit_RDLoss_6081673691437
// MI455X (gfx1250) — compile-verified
//
#include <hip/hip_runtime.h>
#include <math.h>

typedef __attribute__((ext_vector_type(2))) float v2f;
typedef __attribute__((ext_vector_type(8))) float v8f;

#define KT 128          // K columns per LDS tile
#define LSTRIDE 132     // padded LDS row stride (132 mod 64 = 4 -> conflict-free B64 frag reads)
#define SPAN 2048       // K columns handled per block
#define NMAXCHUNK 64

// ---------------------------------------------------------------------------
// Pass 1a: per-row partial max. grid = (chunks, 32 row-tensors). r<16 -> X rows.
// ---------------------------------------------------------------------------
__global__ __launch_bounds__(256) void rowmax_partial_kernel(
    const float* __restrict__ X, const float* __restrict__ Y, int M,
    float* __restrict__ partial) {
  const int r = blockIdx.y;
  const float* src = (r < 16) ? (X + (size_t)r * M) : (Y + (size_t)(r - 16) * M);
  const int chunkLen = M / gridDim.x;
  const float4* base = (const float4*)(src + (size_t)blockIdx.x * chunkLen);
  const int n4 = chunkLen >> 2;
  float m = -3.402823466e38f;
  for (int q = threadIdx.x; q < n4; q += 256) {
    float4 v = base[q];
    m = fmaxf(m, fmaxf(fmaxf(v.x, v.y), fmaxf(v.z, v.w)));
  }
  __shared__ float sm[256];
  sm[threadIdx.x] = m;
  __syncthreads();
  for (int s = 128; s > 0; s >>= 1) {
    if (threadIdx.x < s) sm[threadIdx.x] = fmaxf(sm[threadIdx.x], sm[threadIdx.x + s]);
    __syncthreads();
  }
  if (threadIdx.x == 0) partial[r * gridDim.x + blockIdx.x] = sm[0];
}

// Pass 1b: finalize 32 row maxima (one block, 32 threads)
__global__ void rowmax_final_kernel(const float* __restrict__ partial, int nchunks,
                                    float* __restrict__ rowmax) {
  const int r = threadIdx.x;  // 0..31
  float m = -3.402823466e38f;
  for (int c = 0; c < nchunks; ++c) m = fmaxf(m, partial[r * nchunks + c]);
  rowmax[r] = m;
}

// ---------------------------------------------------------------------------
// Pass 2: fused exp + three Gram matrices via V_WMMA_F32_16X16X4_F32.
// Each block covers SPAN columns of K; partial out = [768 gram | 32 rowsum].
// ---------------------------------------------------------------------------
__global__ __launch_bounds__(256) void gram_kernel(
    const float* __restrict__ X, const float* __restrict__ Y, int M,
    const float* __restrict__ rowmax, float* __restrict__ partialOut) {
  __shared__ __align__(16) float ldsX[16 * LSTRIDE];
  __shared__ __align__(16) float ldsY[16 * LSTRIDE];
  __shared__ float gram[3 * 256];
  __shared__ float red[1024];

  const int t = threadIdx.x;
  const int lane = t & 31;
  const int wave = t >> 5;
  const int k0 = blockIdx.x * SPAN;

  // load mapping: thread t handles rows r0 and r0+8, float4 column cv
  const int r0 = t >> 5;   // 0..7
  const int cv = t & 31;   // 0..31 float4's per row (KT/4 = 32)

  const float mx0 = rowmax[r0],      mx1 = rowmax[r0 + 8];
  const float my0 = rowmax[16 + r0], my1 = rowmax[16 + r0 + 8];

  float sx0 = 0.f, sx1 = 0.f, sy0 = 0.f, sy1 = 0.f;
  v8f gxx = {}, gxy = {}, gyy = {};

  // WMMA fragment addressing (A 16x4 f32 layout == B 4x16 layout per lane)
  const int frow = lane & 15;
  const int koff = (lane >> 4) << 1;

  for (int tile = 0; tile < SPAN / KT; ++tile) {
    const int kc = k0 + tile * KT + cv * 4;
    const float4 x0 = *(const float4*)(X + (size_t)r0 * M + kc);
    const float4 x1 = *(const float4*)(X + (size_t)(r0 + 8) * M + kc);
    const float4 y0 = *(const float4*)(Y + (size_t)r0 * M + kc);
    const float4 y1 = *(const float4*)(Y + (size_t)(r0 + 8) * M + kc);

    __syncthreads();  // previous tile's fragment reads complete

    float4 e;
    e.x = __expf(x0.x - mx0); e.y = __expf(x0.y - mx0);
    e.z = __expf(x0.z - mx0); e.w = __expf(x0.w - mx0);
    *(float4*)&ldsX[r0 * LSTRIDE + cv * 4] = e;
    sx0 += e.x + e.y + e.z + e.w;

    e.x = __expf(x1.x - mx1); e.y = __expf(x1.y - mx1);
    e.z = __expf(x1.z - mx1); e.w = __expf(x1.w - mx1);
    *(float4*)&ldsX[(r0 + 8) * LSTRIDE + cv * 4] = e;
    sx1 += e.x + e.y + e.z + e.w;

    e.x = __expf(y0.x - my0); e.y = __expf(y0.y - my0);
    e.z = __expf(y0.z - my0); e.w = __expf(y0.w - my0);
    *(float4*)&ldsY[r0 * LSTRIDE + cv * 4] = e;
    sy0 += e.x + e.y + e.z + e.w;

    e.x = __expf(y1.x - my1); e.y = __expf(y1.y - my1);
    e.z = __expf(y1.z - my1); e.w = __expf(y1.w - my1);
    *(float4*)&ldsY[(r0 + 8) * LSTRIDE + cv * 4] = e;
    sy1 += e.x + e.y + e.z + e.w;

    __syncthreads();

    // each wave owns K-slices wave, wave+8, wave+16, wave+24 (uniform -> EXEC all 1s)
    for (int s = wave; s < KT / 4; s += 8) {
      const int cb = s * 4 + koff;
      const v2f ax = *(const v2f*)&ldsX[frow * LSTRIDE + cb];
      const v2f ay = *(const v2f*)&ldsY[frow * LSTRIDE + cb];
      gxx = __builtin_amdgcn_wmma_f32_16x16x4_f32(false, ax, false, ax, (short)0, gxx, false, false);
      gxy = __builtin_amdgcn_wmma_f32_16x16x4_f32(false, ax, false, ay, (short)0, gxy, false, false);
      gyy = __builtin_amdgcn_wmma_f32_16x16x4_f32(false, ay, false, ay, (short)0, gyy, false, false);
    }
  }

  __syncthreads();

  // deterministic wave-serial combine of the 8 wave accumulators
  const int rb = (lane >> 4) << 3;   // C/D layout: row = vgpr + 8*(lane/16)
  const int col = lane & 15;
  for (int w = 0; w < 8; ++w) {
    if (wave == w) {
      for (int v = 0; v < 8; ++v) {
        const int idx = (rb + v) * 16 + col;
        if (w == 0) {
          gram[idx] = gxx[v]; gram[256 + idx] = gxy[v]; gram[512 + idx] = gyy[v];
        } else {
          gram[idx] += gxx[v]; gram[256 + idx] += gxy[v]; gram[512 + idx] += gyy[v];
        }
      }
    }
    __syncthreads();
  }

  // row-sum reduction (rows tied statically to thread groups of 32)
  red[t] = sx0; red[256 + t] = sx1; red[512 + t] = sy0; red[768 + t] = sy1;
  __syncthreads();

  float* out = partialOut + (size_t)blockIdx.x * 800;
  out[t] = gram[t]; out[256 + t] = gram[256 + t]; out[512 + t] = gram[512 + t];
  if (t < 32) {
    const int r = t & 15;
    const float* base = red + ((t >= 16) ? 512 : 0) + ((r < 8) ? 32 * r : 256 + 32 * (r - 8));
    float s = 0.f;
    for (int q = 0; q < 32; ++q) s += base[q];
    out[768 + t] = s;
  }
}

// Pass 3: reduce block partials -> final Grams + row sums (one block / pair)
__global__ __launch_bounds__(256) void gram_reduce_kernel(
    const float* __restrict__ partial, int nblocks,
    float* __restrict__ gramOut, float* __restrict__ rowsumOut) {
  const int t = threadIdx.x;
  float s0 = 0.f, s1 = 0.f, s2 = 0.f, rs = 0.f;
  for (int b = 0; b < nblocks; ++b) {
    const float* p = partial + (size_t)b * 800;
    s0 += p[t]; s1 += p[256 + t]; s2 += p[512 + t];
    if (t < 32) rs += p[768 + t];
  }
  gramOut[t] = s0; gramOut[256 + t] = s1; gramOut[512 + t] = s2;
  if (t < 32) rowsumOut[t] = rs;
}

// ---------------------------------------------------------------------------
// Pass 4: Sinkhorn scan on the 16x16 matrices (single block, 256 threads)
// ---------------------------------------------------------------------------
__device__ __forceinline__ float lse16(float v) {
  float m = v;
  for (int d = 1; d < 16; d <<= 1) m = fmaxf(m, __shfl_xor(m, d, 32));
  float s = expf(v - m);
  for (int d = 1; d < 16; d <<= 1) s += __shfl_xor(s, d, 32);
  return m + logf(s);
}

__global__ __launch_bounds__(256) void sinkhorn_kernel(
    const float* __restrict__ G0, const float* __restrict__ R0,
    const float* __restrict__ G1, const float* __restrict__ R1,
    const float* __restrict__ G2, const float* __restrict__ R2,
    float* __restrict__ out) {
  __shared__ float Cxy[256], Cxx[256], Cyy[256];
  __shared__ float f[16], g[16], a[16], b[16];
  __shared__ float nx[16], ny[16];
  __shared__ float acc[32];
  __shared__ float total;

  const int t = threadIdx.x;
  const int i = t >> 4, j = t & 15;
  const float logw = -logf(16.0f);
  if (t == 0) total = 0.0f;

  const float* GP[3] = {G0, G1, G2};
  const float* RP[3] = {R0, R1, R2};

  for (int p = 0; p < 3; ++p) {
    const float* G = GP[p];
    const float* R = RP[p];
    const float sxi = R[i], sxj = R[j], syi = R[16 + i], syj = R[16 + j];
    const float gxx = G[t]        / (sxi * sxj);
    const float gxy = G[256 + t]  / (sxi * syj);
    const float gyy = G[512 + t]  / (syi * syj);

    __syncthreads();
    if (i == j) { nx[i] = gxx; ny[i] = gyy; }
    __syncthreads();

    Cxy[t] = 0.5f * (nx[i] + ny[j]) - gxy;
    Cxx[t] = 0.5f * (nx[i] + nx[j]) - gxx;
    Cyy[t] = 0.5f * (ny[i] + ny[j]) - gyy;
    if (j == 0) { f[i] = 0.f; g[i] = 0.f; a[i] = 0.f; b[i] = 0.f; }
    __syncthreads();

    float e = 1.0f;  // (0.95^2)^k schedule, clamped at blur^2 = 0.0025
    for (int k = 0; k <= 60; ++k) {
      const float eps = (k < 60) ? fmaxf(e, 0.0025f) : 0.0025f;
      const float ft = -eps * lse16(logw + (g[j] - Cxy[t]) / eps);
      const float gt = -eps * lse16(logw + (f[j] - Cxy[j * 16 + i]) / eps);  // Cyx = Cxy^T
      const float at = -eps * lse16(logw + (a[j] - Cxx[t]) / eps);
      const float bt = -eps * lse16(logw + (b[j] - Cyy[t]) / eps);
      __syncthreads();
      if (j == 0) {
        if (k < 60) {
          f[i] = 0.5f * (f[i] + ft); g[i] = 0.5f * (g[i] + gt);
          a[i] = 0.5f * (a[i] + at); b[i] = 0.5f * (b[i] + bt);
        } else {
          acc[i] = ft - at; acc[16 + i] = gt - bt;
        }
      }
      __syncthreads();
      e *= 0.9025f;
    }
    if (t == 0) {
      float s = 0.f;
      for (int q = 0; q < 32; ++q) s += acc[q];
      total += s / 16.0f;   // mean over B=16 of (f-a) plus mean of (g-b)
    }
    __syncthreads();
  }
  if (t == 0) out[0] = total / 3.0f;
}

// ---------------------------------------------------------------------------
extern "C" void kernel_launch(void* const* d_in, const int* in_sizes, int n_in,
                              void* d_out, int out_size, void* d_ws, size_t ws_size,
                              hipStream_t stream) {
  (void)n_in; (void)out_size; (void)ws_size;

  // Dict order: teacher_0, rec_0, teacher_1, rec_1, teacher_2, rec_2
  int Ms[3], GB[3];
  float* maxPartial[3]; float* rowmax[3]; float* gramPartial[3];
  float* gramFinal[3];  float* rowsum[3];

  float* w = (float*)d_ws;
  for (int p = 0; p < 3; ++p) {
    Ms[p] = in_sizes[2 * p] / 16;      // B = 16 rows
    GB[p] = Ms[p] / SPAN;              // 512 / 256 / 128 blocks
    maxPartial[p]  = w; w += 32 * NMAXCHUNK;
    rowmax[p]      = w; w += 32;
    gramPartial[p] = w; w += (size_t)GB[p] * 800;
    gramFinal[p]   = w; w += 768;
    rowsum[p]      = w; w += 32;
  }

  for (int p = 0; p < 3; ++p) {
    const float* X = (const float*)d_in[2 * p];
    const float* Y = (const float*)d_in[2 * p + 1];
    rowmax_partial_kernel<<<dim3(NMAXCHUNK, 32), 256, 0, stream>>>(X, Y, Ms[p], maxPartial[p]);
    rowmax_final_kernel<<<1, 32, 0, stream>>>(maxPartial[p], NMAXCHUNK, rowmax[p]);
    gram_kernel<<<GB[p], 256, 0, stream>>>(X, Y, Ms[p], rowmax[p], gramPartial[p]);
    gram_reduce_kernel<<<1, 256, 0, stream>>>(gramPartial[p], GB[p], gramFinal[p], rowsum[p]);
  }
  sinkhorn_kernel<<<1, 256, 0, stream>>>(gramFinal[0], rowsum[0],
                                         gramFinal[1], rowsum[1],
                                         gramFinal[2], rowsum[2], (float*)d_out);
}